// region_5841155523216
// MI455X (gfx1250) — compile-verified
//
#include <hip/hip_runtime.h>
#include <stdint.h>

#define CLASSES 80
#define CH      85          // channels per anchor in the input
#define OCH     83          // channels per anchor in the output (2 xy + 1 obj + 80 cls)
#define HW      5776        // 76*76
#define TP      128         // pixels (columns) per tile == blockDim.x
#define NBA     80          // batch(16) * anchors(5)

typedef unsigned int u32x4 __attribute__((ext_vector_type(4)));
typedef int          i32x8 __attribute__((ext_vector_type(8)));
typedef int          i32x4 __attribute__((ext_vector_type(4)));

#if defined(__has_include)
#if __has_include(<hip/amd_detail/amd_gfx1250_TDM.h>)
#define REGION_TDM_6ARG 1
#endif
#endif

__device__ __forceinline__ float fast_sigmoid(float v) {
    // v_exp_f32 + v_rcp_f32 path
    return __builtin_amdgcn_rcpf(1.0f + __expf(-v));
}

__global__ __launch_bounds__(TP) void region_head_kernel(const float* __restrict__ x,
                                                         float* __restrict__ out) {
    __shared__ float lds[CH * TP];          // 85 rows x 128 cols, row-major, 43,520 B

    const int tile       = blockIdx.x;      // which group of 128 pixels
    const int ba         = blockIdx.y;      // b*5 + anchor
    const int t          = threadIdx.x;     // column within tile
    const int tile_start = tile * TP;

    // element offset of (channel 0, first column of this tile) in the input
    const size_t in_base = (size_t)ba * CH * HW + (size_t)tile_start;

#if __has_builtin(__builtin_amdgcn_tensor_load_to_lds) && __has_builtin(__builtin_amdgcn_s_wait_tensorcnt)
    // ---- Tensor Data Mover: DMA an 85 x 128 fp32 tile (row stride HW) into LDS ----
    if ((threadIdx.x >> 5) == 0) {          // only wave 0 issues the TDM op
        const unsigned lds_base =
            (unsigned)(unsigned long long)(uintptr_t)(void*)&lds[0]; // low 32 bits = LDS byte offset
        const unsigned long long gaddr = (unsigned long long)(uintptr_t)(x + in_base);

        const unsigned td0 = (unsigned)(HW - tile_start);   // remaining columns -> OOB cols zero-fill
        const unsigned td1 = (unsigned)CH;                  // 85 rows
        const unsigned s0  = (unsigned)HW;                  // row stride (elements)
        const unsigned long long s1 = (unsigned long long)HW * CH; // dim1 stride (unused for 2D)

        // D# group 0 (128b): count=1 | lds_addr | global_addr[56:0] | type=2
        u32x4 g0;
        g0[0] = 1u;
        g0[1] = lds_base;
        g0[2] = (unsigned)(gaddr & 0xFFFFFFFFull);
        g0[3] = ((unsigned)((gaddr >> 32) & 0x01FFFFFFull)) | (2u << 30);

        // D# group 1 (256b): wg_mask=0, data_size=4B, dims/strides/tile dims
        i32x8 g1;
        g1[0] = (int)(2u << 16);                                        // data_size = 4 bytes
        g1[1] = (int)((td0 & 0xFFFFu) << 16);                           // tensor_dim0[15:0]
        g1[2] = (int)(((td0 >> 16) & 0xFFFFu) | ((td1 & 0xFFFFu) << 16)); // td0 hi | tensor_dim1 lo
        g1[3] = (int)(((td1 >> 16) & 0xFFFFu) | ((unsigned)TP << 16));  // td1 hi | tile_dim0 = 128
        g1[4] = (int)((unsigned)CH);                                    // tile_dim1 = 85, tile_dim2 = 0
        g1[5] = (int)s0;                                                // tensor_dim0_stride[31:0]
        g1[6] = (int)(((unsigned)(s1 & 0xFFFFull)) << 16);              // stride0[47:32]=0 | stride1 lo
        g1[7] = (int)((unsigned)((s1 >> 16) & 0xFFFFFFFFull));          // stride1[47:16]

        i32x4 g2 = {0, 0, 0, 0};   // 2D tensor: groups 2/3 unused
        i32x4 g3 = {0, 0, 0, 0};

#ifdef REGION_TDM_6ARG
        i32x8 g1b = {0, 0, 0, 0, 0, 0, 0, 0};
        __builtin_amdgcn_tensor_load_to_lds(g0, g1, g2, g3, g1b, 0);
#else
        __builtin_amdgcn_tensor_load_to_lds(g0, g1, g2, g3, 0);
#endif
        __builtin_amdgcn_s_wait_tensorcnt(0);
    }
#else
    // ---- fallback: cooperative coalesced staging ----
    for (int i = t; i < CH * TP; i += TP) {
        const int c   = i >> 7;          // i / TP
        const int col = i & (TP - 1);    // i % TP
        const int p   = tile_start + col;
        lds[i] = (p < HW) ? x[in_base + (size_t)c * HW + (size_t)col] : 0.0f;
    }
#endif
    __syncthreads();

    const int p = tile_start + t;
    if (p < HW) {
        float* col = &lds[t];            // this thread's private column (stride TP floats)

        // pass 1: running max over the 80 class logits
        float m = -3.0e38f;
        #pragma unroll
        for (int c = 0; c < CLASSES; ++c) {
            m = fmaxf(m, col[(5 + c) * TP]);
        }

        // pass 2: exp(v - m), written back in place, accumulate sum
        float s = 0.0f;
        #pragma unroll
        for (int c = 0; c < CLASSES; ++c) {
            const float e = __expf(col[(5 + c) * TP] - m);
            col[(5 + c) * TP] = e;
            s += e;
        }
        const float inv = __builtin_amdgcn_rcpf(s);

        // output base: channel (a*83 + oc) => flat (ba*83 + oc)*HW + p
        const size_t ob = (size_t)ba * OCH * HW + (size_t)p;
        out[ob + 0 * (size_t)HW] = fast_sigmoid(col[0 * TP]);   // tx
        out[ob + 1 * (size_t)HW] = fast_sigmoid(col[1 * TP]);   // ty
        out[ob + 2 * (size_t)HW] = fast_sigmoid(col[4 * TP]);   // objectness

        // pass 3: normalized class probabilities
        #pragma unroll
        for (int c = 0; c < CLASSES; ++c) {
            out[ob + (size_t)(3 + c) * HW] = col[(5 + c) * TP] * inv;
        }
    }
}

extern "C" void kernel_launch(void* const* d_in, const int* in_sizes, int n_in,
                              void* d_out, int out_size, void* d_ws, size_t ws_size,
                              hipStream_t stream) {
    (void)in_sizes; (void)n_in; (void)d_ws; (void)ws_size; (void)out_size;
    const float* x = (const float*)d_in[0];
    float* out     = (float*)d_out;

    dim3 grid((HW + TP - 1) / TP, NBA);   // 46 x 80 blocks
    dim3 block(TP);                        // 128 threads = 4 waves (wave32)
    hipLaunchKernelGGL(region_head_kernel, grid, block, 0, stream, x, out);
}